// NTfm3D_48833778155660
// MI455X (gfx1250) — compile-verified
//
#include <hip/hip_runtime.h>

typedef __attribute__((ext_vector_type(2))) float v2f;
typedef __attribute__((ext_vector_type(8))) float v8f;

#define HW_    (480 * 640)     // pixels per (b, plane)
#define TILES_ (HW_ / 16)      // 16-pixel tiles per batch

// out[b,i,p] = sum_k m[b,k,p] * ( R[b,k,i,:] . P[b,:,p] + t[b,k,i] )
// Recast as GEMM: A(16x32) x X(32x16pix) with
//   A[i, 4k+j] = transforms[b,k,i,j]   (rows 3..15 zero)
//   X[4k+j, p] = m[k,p] * {px,py,pz,1}[j]
// accumulated via 8x V_WMMA_F32_16X16X4_F32 per 16-pixel tile.
__global__ __launch_bounds__(256) void ntfm3d_wmma_kernel(
    const float* __restrict__ points,      // (B,3,H,W)
    const float* __restrict__ masks,       // (B,8,H,W)
    const float* __restrict__ transforms,  // (B,8,3,4)
    float* __restrict__ out)               // (B,3,H,W)
{
    const int  b    = blockIdx.y;
    const int  lane = threadIdx.x & 31;
    const bool hi   = lane >= 16;          // half-wave: K-slice rows {2,3}
    const int  row  = lane & 15;           // A-matrix row / pixel-in-tile (B column)

    const int wavesPerBlock = blockDim.x >> 5;
    const int wave   = blockIdx.x * wavesPerBlock + (threadIdx.x >> 5);
    const int nwaves = gridDim.x * wavesPerBlock;

    // ---- A operand registers: per K-slice kk, this lane's two A values ----
    // low lanes: A[row, 4kk+0], A[row, 4kk+1]   (R cols 0,1)
    // high lanes: A[row, 4kk+2], A[row, 4kk+3]  (R col 2, t)
    const float* T  = transforms + (size_t)b * (8 * 12);
    const int    j0 = hi ? 2 : 0;
    float a0[8], a1[8];
#pragma unroll
    for (int kk = 0; kk < 8; ++kk) {
        float x0 = 0.0f, x1 = 0.0f;
        if (row < 3) {                      // rows 3..15 of A are zero padding
            x0 = T[kk * 12 + row * 4 + j0];
            x1 = T[kk * 12 + row * 4 + j0 + 1];
        }
        a0[kk] = x0;
        a1[kk] = x1;
    }

    const float* Pb = points + (size_t)b * 3 * HW_;
    const float* Mb = masks  + (size_t)b * 8 * HW_;
    float*       Ob = out    + (size_t)b * 3 * HW_;

    for (int t = wave; t < TILES_; t += nwaves) {
        const int p = t * 16 + row;        // this lane's pixel (= B column N)

        const float px = Pb[p];
        const float py = Pb[HW_ + p];
        const float pz = Pb[2 * HW_ + p];

        float m[8];
#pragma unroll
        for (int kk = 0; kk < 8; ++kk) m[kk] = Mb[(size_t)kk * HW_ + p];

        // B operand selectors: low half feeds X rows {4k+0,4k+1} = m*{px,py},
        // high half feeds X rows {4k+2,4k+3} = m*{pz,1}.
        const float sa = hi ? pz : px;
        const float sb = hi ? 1.0f : py;

        v8f c = {0.f, 0.f, 0.f, 0.f, 0.f, 0.f, 0.f, 0.f};
#pragma unroll
        for (int kk = 0; kk < 8; ++kk) {
            v2f av; av.x = a0[kk];      av.y = a1[kk];
            v2f bv; bv.x = m[kk] * sa;  bv.y = m[kk] * sb;
            // 8 args: (neg_a, A, neg_b, B, c_mod, C, reuse_a, reuse_b)
            c = __builtin_amdgcn_wmma_f32_16x16x4_f32(
                    false, av, false, bv, (short)0, c, false, false);
        }

        // D layout: lanes 0-15, VGPR r holds D[M=r, N=lane]; rows 0..2 are out.
        if (!hi) {
            Ob[p]            = c[0];
            Ob[HW_ + p]      = c[1];
            Ob[2 * HW_ + p]  = c[2];
        }
    }
}

extern "C" void kernel_launch(void* const* d_in, const int* in_sizes, int n_in,
                              void* d_out, int out_size, void* d_ws, size_t ws_size,
                              hipStream_t stream) {
    const float* points     = (const float*)d_in[0];
    const float* masks      = (const float*)d_in[1];
    const float* transforms = (const float*)d_in[2];
    float*       out        = (float*)d_out;

    // 240 blocks x 8 waves = 1920 waves per batch -> exactly 10 tiles/wave
    dim3 block(256, 1, 1);
    dim3 grid(240, 16, 1);
    hipLaunchKernelGGL(ntfm3d_wmma_kernel, grid, block, 0, stream,
                       points, masks, transforms, out);
}